// QuanvolutionDual_65481071410456
// MI455X (gfx1250) — compile-verified
//
#include <hip/hip_runtime.h>
#include <math.h>

typedef __attribute__((ext_vector_type(2))) float v2f;
typedef __attribute__((ext_vector_type(8))) float v8f;

#define FEAT     1568
#define FSTRIDE  1569           // odd stride -> 16 rows hit 16 distinct LDS banks
#define TILE     16             // batch rows per block (one WMMA D tile)
#define NWAVES   8              // 256 threads / wave32
#define KPW      (FEAT / NWAVES) // 196 K-elements per wave

__global__ __launch_bounds__(256)
void quanv_fused_kernel(const float* __restrict__ x,       // [B,1,28,28]
                        const float* __restrict__ conv_w,  // [4,1,2,2]
                        const float* __restrict__ conv_b,  // [4]
                        const float* __restrict__ fc_w,    // [10,1568]
                        const float* __restrict__ fc_b,    // [10]
                        float* __restrict__ out)           // [B,10] log_softmax
{
    __shared__ float feat[TILE * FSTRIDE];          // ~100 KB: fused feature tile
    __shared__ float cpart[NWAVES * TILE * 16];     // 8 KB: per-wave partial C

    const int t    = threadIdx.x;
    const int row0 = blockIdx.x * TILE;

    // ---- hoist tiny uniform weights ----
    float cw[16], cb[4];
#pragma unroll
    for (int q = 0; q < 16; ++q) cw[q] = conv_w[q];
#pragma unroll
    for (int q = 0; q < 4; ++q)  cb[q] = conv_b[q];

    // =====================================================================
    // Phase 1: fused features (conv 2x2/s2 + 4-qubit quanvolution) -> LDS
    // feat[r][c*196 + patch]            (conv part,   c = 0..3)
    // feat[r][784 + 4*patch + w]        (quantum part, w = 0..3)
    // =====================================================================
    const float CO = 0.9238795325112867f;  // cos(pi/8)
    const float SI = 0.3826834323650898f;  // sin(pi/8)

    for (int task = t; task < TILE * 196; task += 256) {
        const int r  = task / 196;
        const int pp = task - r * 196;       // patch index i*14+j
        const int i  = pp / 14;
        const int j  = pp - i * 14;

        const float* xp = x + (size_t)(row0 + r) * 784 + (2 * i) * 28 + 2 * j;
        const float p0 = xp[0],  p1 = xp[1];
        const float p2 = xp[28], p3 = xp[29];

        float* fr = feat + r * FSTRIDE;

        // ---- conv branch ----
#pragma unroll
        for (int c = 0; c < 4; ++c) {
            fr[c * 196 + pp] = cw[4*c+0]*p0 + cw[4*c+1]*p1 +
                               cw[4*c+2]*p2 + cw[4*c+3]*p3 + cb[c];
        }

        // ---- quantum branch: RY(p_k)|0> product state, index q0q1q2q3 ----
        float s0, c0, s1, c1, s2, c2, s3, c3;
        __sincosf(0.5f * p0, &s0, &c0);
        __sincosf(0.5f * p1, &s1, &c1);
        __sincosf(0.5f * p2, &s2, &c2);
        __sincosf(0.5f * p3, &s3, &c3);

        float st[16];
#pragma unroll
        for (int q = 0; q < 16; ++q) {
            st[q] = ((q & 8) ? s0 : c0) * ((q & 4) ? s1 : c1) *
                    ((q & 2) ? s2 : c2) * ((q & 1) ? s3 : c3);
        }

#pragma unroll
        for (int layer = 0; layer < 2; ++layer) {
            // CNOT(0,1): in control=1 subspace swap target bit
#pragma unroll
            for (int q = 0; q < 16; ++q)
                if ((q & 8) && !(q & 4)) { float tm = st[q]; st[q] = st[q | 4]; st[q | 4] = tm; }
            // CNOT(1,2)
#pragma unroll
            for (int q = 0; q < 16; ++q)
                if ((q & 4) && !(q & 2)) { float tm = st[q]; st[q] = st[q | 2]; st[q | 2] = tm; }
            // CNOT(2,3)
#pragma unroll
            for (int q = 0; q < 16; ++q)
                if ((q & 2) && !(q & 1)) { float tm = st[q]; st[q] = st[q | 1]; st[q | 1] = tm; }
            // RY(wire3, pi/4)
#pragma unroll
            for (int q = 0; q < 16; q += 2) {
                const float e = st[q], o = st[q + 1];
                st[q]     = CO * e - SI * o;
                st[q + 1] = SI * e + CO * o;
            }
        }

        float z0 = 0.f, z1 = 0.f, z2 = 0.f, z3 = 0.f;
#pragma unroll
        for (int q = 0; q < 16; ++q) {
            const float pr = st[q] * st[q];
            z0 += (q & 8) ? -pr : pr;
            z1 += (q & 4) ? -pr : pr;
            z2 += (q & 2) ? -pr : pr;
            z3 += (q & 1) ? -pr : pr;
        }
        fr[784 + 4 * pp + 0] = z0;
        fr[784 + 4 * pp + 1] = z1;
        fr[784 + 4 * pp + 2] = z2;
        fr[784 + 4 * pp + 3] = z3;
    }
    __syncthreads();

    // =====================================================================
    // Phase 2: FC GEMM via V_WMMA_F32_16X16X4_F32, K split over 8 waves.
    //  A (16x4 f32): lane L -> M=L%16, VGPR v -> K = 2*(L/16)+v
    //  B (4x16 f32): lane L -> N=L%16, VGPR v -> K = 2*(L/16)+v
    //  C/D          : VGPR r, lane half h -> M = r+8h, N = L%16
    // Padded columns 10..15 compute a duplicate of column 9 (clamped row);
    // their results are finite and discarded in phase 3, so no masking
    // is needed in the hot loop.
    // =====================================================================
    const int lane = t & 31;
    // wave id is wave-uniform: pin it to an SGPR so the K-loop uses scalar
    // control flow (EXEC stays all-1s across every WMMA, per ISA rules).
    const int wv   = __builtin_amdgcn_readfirstlane(t >> 5);
    const int mn   = lane & 15;   // A-row / B-col / D-col
    const int h    = lane >> 4;

    const float* wrow = fc_w + (size_t)((mn < 10) ? mn : 9) * FEAT;

    v8f acc = {};
    const int kbeg = wv * KPW;
#pragma unroll 7
    for (int it = 0; it < KPW / 4; ++it) {
        const int k0 = kbeg + 4 * it + 2 * h;
        v2f a, b;
        a.x = feat[mn * FSTRIDE + k0];
        a.y = feat[mn * FSTRIDE + k0 + 1];
        b.x = wrow[k0];
        b.y = wrow[k0 + 1];
        acc = __builtin_amdgcn_wmma_f32_16x16x4_f32(
            /*neg_a=*/false, a, /*neg_b=*/false, b,
            /*c_mod=*/(short)0, acc, /*reuse_a=*/false, /*reuse_b=*/false);
    }

    float* cp = cpart + wv * (TILE * 16);
#pragma unroll
    for (int r = 0; r < 8; ++r) {
        cp[(r + 8 * h) * 16 + mn] = acc[r];
    }
    __syncthreads();

    // =====================================================================
    // Phase 3: reduce wave partials, add bias, log_softmax over 10 cols.
    // thread t holds row m = t/16, col n = t%16 (matches wave32 layout:
    // lanes 0-15 = one row, lanes 16-31 = next row -> shfl width 16).
    // =====================================================================
    const int mrow = t >> 4;
    const int ncol = t & 15;

    float logit = 0.f;
#pragma unroll
    for (int w = 0; w < NWAVES; ++w) logit += cpart[w * (TILE * 16) + t];
    logit = (ncol < 10) ? (logit + fc_b[ncol]) : -3.0e38f;

    float mx = logit;
#pragma unroll
    for (int off = 8; off > 0; off >>= 1)
        mx = fmaxf(mx, __shfl_xor(mx, off, 16));

    float ex = __expf(logit - mx);        // padded cols -> exp(-huge) = 0
    float sum = ex;
#pragma unroll
    for (int off = 8; off > 0; off >>= 1)
        sum += __shfl_xor(sum, off, 16);

    if (ncol < 10)
        out[(size_t)(row0 + mrow) * 10 + ncol] = logit - mx - __logf(sum);
}

extern "C" void kernel_launch(void* const* d_in, const int* in_sizes, int n_in,
                              void* d_out, int out_size, void* d_ws, size_t ws_size,
                              hipStream_t stream) {
    (void)n_in; (void)out_size; (void)d_ws; (void)ws_size;
    const float* x      = (const float*)d_in[0];
    const float* conv_w = (const float*)d_in[1];
    const float* conv_b = (const float*)d_in[2];
    const float* fc_w   = (const float*)d_in[3];
    const float* fc_b   = (const float*)d_in[4];
    float* out = (float*)d_out;

    const int B = in_sizes[0] / 784;       // 8192
    dim3 grid(B / TILE);                   // 512 blocks of 16 rows
    quanv_fused_kernel<<<grid, 256, 0, stream>>>(x, conv_w, conv_b, fc_w, fc_b, out);
}